// _Model_35115652612804
// MI455X (gfx1250) — compile-verified
//
#include <hip/hip_runtime.h>

typedef float v2f __attribute__((ext_vector_type(2)));
typedef float v8f __attribute__((ext_vector_type(8)));

#define RNN_H 8
#define NLAYERS 8
#define TMAX 64

__device__ __forceinline__ float fast_tanh(float x) {
  // tanh(x) = 1 - 2/(exp(2x)+1) ; v_exp_f32 + v_rcp_f32
  float e = __expf(2.0f * x);
  float r = __builtin_amdgcn_rcpf(e + 1.0f);
  return __builtin_fmaf(-2.0f, r, 1.0f);
}

// Build a 4x16 B-matrix fragment for V_WMMA_F32_16X16X4_F32.
// D[m][n] = sum_k A[m][k]*B[k][n]; we want B[k][n] = W[n][k0+k] (W row-major [8][Ktot]).
// Layout: VGPR0 = K (0 | 2 per lane-half), VGPR1 = K (1 | 3); col n = lane%16, n>=8 -> 0.
__device__ __forceinline__ v2f load_B(const float* __restrict__ W, int Ktot, int k0,
                                      int n, int half) {
  v2f r;
  if (n < 8) {
    const float* p = W + n * Ktot + k0 + 2 * half;
    r.x = p[0];
    r.y = p[1];
  } else {
    r.x = 0.0f;
    r.y = 0.0f;
  }
  return r;
}

// One wave runs a full 8-layer RNN for 16 batch rows.
// Sq[t][b][h] is the per-wave LDS sequence buffer (overwritten in place per layer);
// it also implements the D-layout -> A-layout transpose between timesteps.
template <int D>
__device__ void rnn_wave(const float* __restrict__ X,     // [B,T,D]
                         const float* __restrict__ W0,    // [8,D]
                         const float* __restrict__ Wih,   // [L-1,8,8]
                         const float* __restrict__ Whh,   // [L,8,8]
                         const float* __restrict__ bih,   // [L,8]
                         const float* __restrict__ bhh,   // [L,8]
                         float (*Sq)[16][RNN_H],
                         int bb, int T, int lane) {
  const int m = lane & 15;     // A-row / D-col index
  const int half = lane >> 4;  // lane-half selects K pair / row offset +8
  constexpr int NK0 = D / 4;

  // ---- weights as register-resident WMMA B fragments ----
  v2f B0[NK0];
#pragma unroll
  for (int j = 0; j < NK0; ++j) B0[j] = load_B(W0, D, 4 * j, m, half);

  v2f Bhh[NLAYERS][2];
  v2f Bih[NLAYERS - 1][2];
  float biasv[NLAYERS];
#pragma unroll
  for (int l = 0; l < NLAYERS; ++l) {
    Bhh[l][0] = load_B(Whh + l * RNN_H * RNN_H, RNN_H, 0, m, half);
    Bhh[l][1] = load_B(Whh + l * RNN_H * RNN_H, RNN_H, 4, m, half);
    biasv[l] = (m < 8) ? (bih[l * RNN_H + m] + bhh[l * RNN_H + m]) : 0.0f;
  }
#pragma unroll
  for (int l = 0; l < NLAYERS - 1; ++l) {
    Bih[l][0] = load_B(Wih + l * RNN_H * RNN_H, RNN_H, 0, m, half);
    Bih[l][1] = load_B(Wih + l * RNN_H * RNN_H, RNN_H, 4, m, half);
  }

  // ---- layer-0 input projection: M = 16 timesteps of one batch row, K = D ----
  for (int b = 0; b < 16; ++b) {
    const float* xrow = X + (size_t)(bb + b) * T * D;
    for (int t0 = 0; t0 < T; t0 += 16) {
      v8f c;
#pragma unroll
      for (int r = 0; r < 8; ++r) c[r] = biasv[0];
#pragma unroll
      for (int j = 0; j < NK0; ++j) {
        const float* p = xrow + (t0 + m) * D + 4 * j + 2 * half;
        v2f a;
        a.x = p[0];
        a.y = p[1];
        c = __builtin_amdgcn_wmma_f32_16x16x4_f32(false, a, false, B0[j],
                                                  (short)0, c, false, false);
      }
      if (m < 8) {
#pragma unroll
        for (int r = 0; r < 8; ++r) Sq[t0 + r + 8 * half][b][m] = c[r];
      }
    }
  }

  // ---- recurrence, layer by layer, in place in Sq ----
#pragma unroll
  for (int layer = 0; layer < NLAYERS; ++layer) {
    for (int t = 0; t < T; ++t) {
      v8f c;
      if (layer == 0) {
        // C = precomputed xw_t (+bias); padded cols -> 0
#pragma unroll
        for (int r = 0; r < 8; ++r)
          c[r] = (m < 8) ? Sq[t][r + 8 * half][m] : 0.0f;
      } else {
#pragma unroll
        for (int r = 0; r < 8; ++r) c[r] = biasv[layer];
        // input projection from previous layer's output at step t (A layout)
        const float* p0 = &Sq[t][m][2 * half];
        v2f a0, a1;
        a0.x = p0[0]; a0.y = p0[1];
        a1.x = p0[4]; a1.y = p0[5];
        c = __builtin_amdgcn_wmma_f32_16x16x4_f32(false, a0, false, Bih[layer - 1][0],
                                                  (short)0, c, false, false);
        c = __builtin_amdgcn_wmma_f32_16x16x4_f32(false, a1, false, Bih[layer - 1][1],
                                                  (short)0, c, false, false);
      }
      if (t > 0) {
        // h_{t-1} of this layer was written to Sq[t-1] on the previous step
        const float* ph = &Sq[t - 1][m][2 * half];
        v2f h0, h1;
        h0.x = ph[0]; h0.y = ph[1];
        h1.x = ph[4]; h1.y = ph[5];
        c = __builtin_amdgcn_wmma_f32_16x16x4_f32(false, h0, false, Bhh[layer][0],
                                                  (short)0, c, false, false);
        c = __builtin_amdgcn_wmma_f32_16x16x4_f32(false, h1, false, Bhh[layer][1],
                                                  (short)0, c, false, false);
      }
#pragma unroll
      for (int r = 0; r < 8; ++r) c[r] = fast_tanh(c[r]);
      if (m < 8) {
#pragma unroll
        for (int r = 0; r < 8; ++r) Sq[t][r + 8 * half][m] = c[r];
      }
    }
  }
}

__global__ __launch_bounds__(64) void fused_rnn_mlp(
    const float* __restrict__ Xas, const float* __restrict__ Xlink,
    const float* __restrict__ Xpath,
    const float* __restrict__ as_w0, const float* __restrict__ as_wih,
    const float* __restrict__ as_whh, const float* __restrict__ as_bih,
    const float* __restrict__ as_bhh,
    const float* __restrict__ lk_w0, const float* __restrict__ lk_wih,
    const float* __restrict__ lk_whh, const float* __restrict__ lk_bih,
    const float* __restrict__ lk_bhh,
    const float* __restrict__ path_w, const float* __restrict__ path_b,
    const float* __restrict__ w1, const float* __restrict__ b1,
    const float* __restrict__ w2, const float* __restrict__ b2,
    const float* __restrict__ w3, const float* __restrict__ b3,
    const float* __restrict__ w4, const float* __restrict__ b4,
    float* __restrict__ out, int T) {
  __shared__ float S[2][TMAX][16][RNN_H];  // 64 KB: one 32 KB seq buffer per wave

  const int tid = threadIdx.x;
  const int wave = tid >> 5;
  const int lane = tid & 31;
  const int bb = blockIdx.x * 16;

  if (wave == 0)
    rnn_wave<32>(Xas, as_w0, as_wih, as_whh, as_bih, as_bhh, S[0], bb, T, lane);
  else
    rnn_wave<16>(Xlink, lk_w0, lk_wih, lk_whh, lk_bih, lk_bhh, S[1], bb, T, lane);

  __syncthreads();

  // ---- fused MLP head: one lane per batch row ----
  if (tid < 16) {
    const int b = bb + tid;
    float z[24];
#pragma unroll
    for (int i = 0; i < 8; ++i) z[i] = S[0][T - 1][tid][i];
#pragma unroll
    for (int i = 0; i < 8; ++i) z[8 + i] = S[1][T - 1][tid][i];
#pragma unroll
    for (int n = 0; n < 8; ++n) {
      float acc = path_b[n];
#pragma unroll
      for (int d = 0; d < 16; ++d)
        acc = __builtin_fmaf(Xpath[b * 16 + d], path_w[n * 16 + d], acc);
      z[16 + n] = fmaxf(acc, 0.0f);
    }
    float z1[32];
#pragma unroll
    for (int j = 0; j < 32; ++j) {
      float a = b1[j];
#pragma unroll
      for (int i = 0; i < 24; ++i) a = __builtin_fmaf(w1[j * 24 + i], z[i], a);
      z1[j] = fmaxf(a, 0.0f);
    }
    float z2[16];
#pragma unroll
    for (int j = 0; j < 16; ++j) {
      float a = b2[j];
#pragma unroll
      for (int i = 0; i < 32; ++i) a = __builtin_fmaf(w2[j * 32 + i], z1[i], a);
      z2[j] = fmaxf(a, 0.0f);
    }
    float z3[8];
#pragma unroll
    for (int j = 0; j < 8; ++j) {
      float a = b3[j];
#pragma unroll
      for (int i = 0; i < 16; ++i) a = __builtin_fmaf(w3[j * 16 + i], z2[i], a);
      z3[j] = fmaxf(a, 0.0f);
    }
    float o = b4[0];
#pragma unroll
    for (int i = 0; i < 8; ++i) o = __builtin_fmaf(w4[i], z3[i], o);
    out[b] = o;
  }
}

extern "C" void kernel_launch(void* const* d_in, const int* in_sizes, int n_in,
                              void* d_out, int out_size, void* d_ws, size_t ws_size,
                              hipStream_t stream) {
  (void)n_in; (void)d_ws; (void)ws_size; (void)out_size;
  const int B = in_sizes[2] / 16;        // X_path_level is [B,16]
  const int T = in_sizes[0] / (B * 32);  // X_as_level is [B,T,32]

  fused_rnn_mlp<<<B / 16, 64, 0, stream>>>(
      (const float*)d_in[0], (const float*)d_in[1], (const float*)d_in[2],
      (const float*)d_in[3], (const float*)d_in[4], (const float*)d_in[5],
      (const float*)d_in[6], (const float*)d_in[7],
      (const float*)d_in[8], (const float*)d_in[9], (const float*)d_in[10],
      (const float*)d_in[11], (const float*)d_in[12],
      (const float*)d_in[13], (const float*)d_in[14],
      (const float*)d_in[15], (const float*)d_in[16],
      (const float*)d_in[17], (const float*)d_in[18],
      (const float*)d_in[19], (const float*)d_in[20],
      (const float*)d_in[21], (const float*)d_in[22],
      (float*)d_out, T);
}